// Decoder_4544075399464
// MI455X (gfx1250) — compile-verified
//
#include <hip/hip_runtime.h>

typedef __attribute__((ext_vector_type(2))) float v2f;
typedef __attribute__((ext_vector_type(8))) float v8f;
typedef int vi4 __attribute__((vector_size(16)));   // matches builtin param type

#define BATCH 128
#define EMB   256
#define HID   1024
#define VOCAB 32000

#define GLOBAL_AS __attribute__((address_space(1)))
#define LDS_AS    __attribute__((address_space(3)))

#if __has_builtin(__builtin_amdgcn_global_load_async_to_lds_b128)
#define HAS_ASYNC_LDS 1
#else
#define HAS_ASYNC_LDS 0
#endif

#if __has_builtin(__builtin_amdgcn_s_wait_asynccnt)
#define WAIT_ASYNC(n) __builtin_amdgcn_s_wait_asynccnt(n)
#else
#define WAIT_ASYNC(n) asm volatile("s_wait_asynccnt %0" ::"i"(n) : "memory")
#endif

__device__ __forceinline__ v8f wmma_f32(v2f a, v2f b, v8f c) {
  // D = A(16x4) * B(4x16) + C(16x16), all f32
  return __builtin_amdgcn_wmma_f32_16x16x4_f32(
      /*neg_a=*/false, a, /*neg_b=*/false, b,
      /*c_mod=*/(short)0, c, /*reuse_a=*/false, /*reuse_b=*/false);
}

#if HAS_ASYNC_LDS
__device__ __forceinline__ GLOBAL_AS vi4* to_global_v4(const void* p) {
  return (GLOBAL_AS vi4*)(void*)p;
}
__device__ __forceinline__ LDS_AS vi4* to_lds_v4(void* p) {
  return (LDS_AS vi4*)p;
}
#endif

// ---------------- embedding gather: e[b][k] = emb[x[b]][k] ----------------
__global__ void __launch_bounds__(256)
gather_kernel(const int* __restrict__ x, const float* __restrict__ emb,
              float* __restrict__ e) {
  int idx = blockIdx.x * 256 + threadIdx.x;   // 128*256 total
  int b = idx >> 8;
  int k = idx & 255;
  e[idx] = emb[(size_t)x[b] * EMB + k];
}

// ---------------- fused GRU layer ----------------
// grid.x = HID/16 (one block per 16-col tile of the hidden dim)
// block  = 256 threads = 8 waves; wave w owns rows [16w, 16w+16)
__global__ void __launch_bounds__(256)
gru_kernel(const float* __restrict__ xin, int Kx,
           const float* __restrict__ h,
           const float* __restrict__ Wi, const float* __restrict__ Wr,
           const float* __restrict__ bi, const float* __restrict__ br,
           float* __restrict__ hn) {
  const int lane  = threadIdx.x & 31;
  const int wave  = threadIdx.x >> 5;
  const int hi    = lane >> 4;       // 0 or 1 (half-wave)
  const int l16   = lane & 15;
  const int mbase = wave * 16;
  const int n0    = blockIdx.x * 16;
  const int row   = mbase + l16;     // A-matrix row for this lane
  const int col   = n0 + l16;        // B/C/D column for this lane

  v8f c_iz = {}, c_ir = {}, c_ih = {};
  v8f c_rz = {}, c_rr = {}, c_rh = {};

  // gi = xin @ Wi   (K = Kx), three gate column-tiles share one A tile
  for (int k = 0; k < Kx; k += 4) {
    v2f a;
    a.x = xin[(size_t)row * Kx + k + 2 * hi];
    a.y = xin[(size_t)row * Kx + k + 2 * hi + 1];
    const float* w = Wi + (size_t)(k + 2 * hi) * (3 * HID) + col;
    v2f b0, b1, b2;
    b0.x = w[0];           b0.y = w[3 * HID];
    b1.x = w[HID];         b1.y = w[3 * HID + HID];
    b2.x = w[2 * HID];     b2.y = w[3 * HID + 2 * HID];
    c_iz = wmma_f32(a, b0, c_iz);
    c_ir = wmma_f32(a, b1, c_ir);
    c_ih = wmma_f32(a, b2, c_ih);
  }

  // gr = h @ Wr   (K = HID)
  for (int k = 0; k < HID; k += 4) {
    v2f a;
    a.x = h[(size_t)row * HID + k + 2 * hi];
    a.y = h[(size_t)row * HID + k + 2 * hi + 1];
    const float* w = Wr + (size_t)(k + 2 * hi) * (3 * HID) + col;
    v2f b0, b1, b2;
    b0.x = w[0];           b0.y = w[3 * HID];
    b1.x = w[HID];         b1.y = w[3 * HID + HID];
    b2.x = w[2 * HID];     b2.y = w[3 * HID + 2 * HID];
    c_rz = wmma_f32(a, b0, c_rz);
    c_rr = wmma_f32(a, b1, c_rr);
    c_rh = wmma_f32(a, b2, c_rh);
  }

  // gate fusion: z = sig(iz+rz), r = sig(ir+rr), hh = tanh(ih + r*rh)
  const float biz = bi[col], bir = bi[col + HID], bih = bi[col + 2 * HID];
  const float brz = br[col], brr = br[col + HID], brh = br[col + 2 * HID];
#pragma unroll
  for (int v = 0; v < 8; ++v) {
    const int m = mbase + hi * 8 + v;   // C/D row for VGPR v (ISA layout)
    float iz = c_iz[v] + biz, ir = c_ir[v] + bir, ih = c_ih[v] + bih;
    float rz = c_rz[v] + brz, rr = c_rr[v] + brr, rh = c_rh[v] + brh;
    float z  = 1.0f / (1.0f + expf(-(iz + rz)));
    float r  = 1.0f / (1.0f + expf(-(ir + rr)));
    float hh = tanhf(ih + r * rh);
    float hp = h[(size_t)m * HID + col];
    hn[(size_t)m * HID + col] = z * hp + (1.0f - z) * hh;
  }
}

// ---------------- output projection: logits = h1n @ ffW + ffb ----------------
// grid.x = VOCAB/32 (32-col blocks => full 128B lines per staged row)
// block  = 256 threads = 8 waves; wave w owns rows [16w,16w+16), 2 N-tiles.
// ffW tiles double-buffered through LDS via async loads (ASYNCcnt).
#define KC   32               // K rows per staged chunk
#define SROW 40               // padded LDS row stride (floats)
#define NCH  (HID / KC)       // 32 chunks

__global__ void __launch_bounds__(256)
ff_kernel(const float* __restrict__ A, const float* __restrict__ W,
          const float* __restrict__ bias, float* __restrict__ out) {
  __shared__ float bs[2][KC][SROW];

  const int tid   = threadIdx.x;
  const int lane  = tid & 31;
  const int wave  = tid >> 5;
  const int hi    = lane >> 4;
  const int l16   = lane & 15;
  const int mbase = wave * 16;
  const int n0    = blockIdx.x * 32;
  const int row   = mbase + l16;

  const int srow = tid >> 3;   // 0..31: staged K row
  const int sseg = tid & 7;    // 0..7 : 16B segment within 128B row

  auto stage = [&](int buf, int chunk) {
    const float* gsrc = W + (size_t)(chunk * KC + srow) * VOCAB + n0 + sseg * 4;
    float* ldst = &bs[buf][srow][sseg * 4];
#if HAS_ASYNC_LDS
    __builtin_amdgcn_global_load_async_to_lds_b128(to_global_v4(gsrc),
                                                   to_lds_v4(ldst),
                                                   /*offset=*/0, /*cpol=*/0);
#else
    *(float4*)ldst = *(const float4*)gsrc;
#endif
  };

  stage(0, 0);
  stage(1, 1);

  v8f acc0 = {}, acc1 = {};
  for (int c = 0; c < NCH; ++c) {
#if HAS_ASYNC_LDS
    if (c < NCH - 1) WAIT_ASYNC(1);   // stage(c) complete (in-order), stage(c+1) may fly
    else             WAIT_ASYNC(0);
#endif
    __syncthreads();                  // chunk c visible to all waves

    const int   buf  = c & 1;
    const float* arow = A + (size_t)row * HID + c * KC + 2 * hi;
#pragma unroll
    for (int kk = 0; kk < KC; kk += 4) {
      v2f a;
      a.x = arow[kk];
      a.y = arow[kk + 1];
      v2f b0, b1;
      b0.x = bs[buf][kk + 2 * hi][l16];
      b0.y = bs[buf][kk + 2 * hi + 1][l16];
      b1.x = bs[buf][kk + 2 * hi][l16 + 16];
      b1.y = bs[buf][kk + 2 * hi + 1][l16 + 16];
      acc0 = wmma_f32(a, b0, acc0);
      acc1 = wmma_f32(a, b1, acc1);
    }

    __syncthreads();                  // all waves done reading buf before overwrite
    if (c + 2 < NCH) stage(buf, c + 2);
  }

  const float b0v = bias[n0 + l16];
  const float b1v = bias[n0 + 16 + l16];
#pragma unroll
  for (int v = 0; v < 8; ++v) {
    const int m = mbase + hi * 8 + v;
    out[(size_t)m * VOCAB + n0 + l16]      = acc0[v] + b0v;
    out[(size_t)m * VOCAB + n0 + 16 + l16] = acc1[v] + b1v;
  }
}

extern "C" void kernel_launch(void* const* d_in, const int* in_sizes, int n_in,
                              void* d_out, int out_size, void* d_ws, size_t ws_size,
                              hipStream_t stream) {
  const int*   x   = (const int*)  d_in[0];
  const float* h0  = (const float*)d_in[1];
  const float* h1  = (const float*)d_in[2];
  const float* emb = (const float*)d_in[3];
  const float* Wi0 = (const float*)d_in[4];
  const float* Wr0 = (const float*)d_in[5];
  const float* bi0 = (const float*)d_in[6];
  const float* br0 = (const float*)d_in[7];
  const float* Wi1 = (const float*)d_in[8];
  const float* Wr1 = (const float*)d_in[9];
  const float* bi1 = (const float*)d_in[10];
  const float* br1 = (const float*)d_in[11];
  const float* ffW = (const float*)d_in[12];
  const float* ffb = (const float*)d_in[13];

  float* logits = (float*)d_out;
  float* h0n    = logits + (size_t)BATCH * VOCAB;
  float* h1n    = h0n    + (size_t)BATCH * HID;
  float* e      = (float*)d_ws;          // 128*256 f32 = 131 KB scratch

  gather_kernel<<<(BATCH * EMB) / 256, 256, 0, stream>>>(x, emb, e);
  gru_kernel<<<HID / 16, 256, 0, stream>>>(e,   EMB, h0, Wi0, Wr0, bi0, br0, h0n);
  gru_kernel<<<HID / 16, 256, 0, stream>>>(h0n, HID, h1, Wi1, Wr1, bi1, br1, h1n);
  ff_kernel<<<VOCAB / 32, 256, 0, stream>>>(h1n, ffW, ffb, logits);
}